// Attn_cat_freq_71090298683718
// MI455X (gfx1250) — compile-verified
//
#include <hip/hip_runtime.h>

// Problem constants (match the reference).
#define NUM_SLOTS 168
#define NUM_CATS  2048   // floats per row -> 8192 bytes per row

// ---------------------------------------------------------------------------
// CDNA5 async global<->LDS helpers (gfx1250, ASYNCcnt-tracked DMA path).
// The async engine's "LDS address" operand is a single VGPR holding the byte
// offset into the workgroup's LDS allocation; the low 32 bits of a generic
// pointer to a __shared__ object are exactly that offset.
// ---------------------------------------------------------------------------
__device__ __forceinline__ unsigned lds_offset_of(const void* p) {
    return (unsigned)(unsigned long long)p;
}

__device__ __forceinline__ void async_load_b128_to_lds(unsigned lds_off, const void* gsrc) {
    // ASYNCcnt++ ; per-lane: copies 16 bytes from global address gsrc to LDS offset lds_off
    asm volatile("global_load_async_to_lds_b128 %0, %1, off"
                 :
                 : "v"(lds_off), "v"(gsrc)
                 : "memory");
}

__device__ __forceinline__ void async_store_b128_from_lds(void* gdst, unsigned lds_off) {
    // ASYNCcnt++ ; per-lane: copies 16 bytes from LDS offset lds_off to global address gdst
    asm volatile("global_store_async_from_lds_b128 %0, %1, off"
                 :
                 : "v"(gdst), "v"(lds_off)
                 : "memory");
}

__device__ __forceinline__ void wait_asynccnt0() {
    asm volatile("s_wait_asynccnt 0" ::: "memory");
}

// ---------------------------------------------------------------------------
// Kernel 1: row softmax of the 168 x 2048 table into the workspace.
// One block per slot row; 256 threads; each thread owns 8 floats (two
// float4s). Block-wide max/sum reductions through LDS. ~1.4 MB of traffic,
// negligible next to the 839 MB gather.
// ---------------------------------------------------------------------------
__global__ __launch_bounds__(256)
void softmax_rows_kernel(const float* __restrict__ M, float* __restrict__ P) {
    const int row = blockIdx.x;           // 0 .. NUM_SLOTS-1
    const int tid = threadIdx.x;          // 0 .. 255

    const float4* src = (const float4*)(M + (size_t)row * NUM_CATS);
    float4*       dst = (float4*)(P + (size_t)row * NUM_CATS);

    float4 a = src[tid];
    float4 b = src[tid + 256];

    // --- block max ---
    float m = fmaxf(fmaxf(fmaxf(a.x, a.y), fmaxf(a.z, a.w)),
                    fmaxf(fmaxf(b.x, b.y), fmaxf(b.z, b.w)));
    __shared__ float red[256];
    red[tid] = m;
    __syncthreads();
    for (int s = 128; s > 0; s >>= 1) {
        if (tid < s) red[tid] = fmaxf(red[tid], red[tid + s]);
        __syncthreads();
    }
    const float rowmax = red[0];
    __syncthreads();

    // --- exp + block sum ---
    a.x = __expf(a.x - rowmax); a.y = __expf(a.y - rowmax);
    a.z = __expf(a.z - rowmax); a.w = __expf(a.w - rowmax);
    b.x = __expf(b.x - rowmax); b.y = __expf(b.y - rowmax);
    b.z = __expf(b.z - rowmax); b.w = __expf(b.w - rowmax);

    float s8 = (a.x + a.y + a.z + a.w) + (b.x + b.y + b.z + b.w);
    red[tid] = s8;
    __syncthreads();
    for (int s = 128; s > 0; s >>= 1) {
        if (tid < s) red[tid] += red[tid + s];
        __syncthreads();
    }
    const float inv = 1.0f / red[0];

    a.x *= inv; a.y *= inv; a.z *= inv; a.w *= inv;
    b.x *= inv; b.y *= inv; b.z *= inv; b.w *= inv;
    dst[tid]       = a;
    dst[tid + 256] = b;
}

// ---------------------------------------------------------------------------
// Kernel 2: the gather, double-buffered through the gfx1250 async DMA path.
// 512 threads/block; lane tid moves bytes [tid*16, tid*16+16) of the 8 KB
// row, so wave w's LDS chunk is wave-private for both directions -> no
// workgroup barrier ever needed.
//
// Pipeline per wave:   L(i) -> buf[i&1]
//                      s_wait_asynccnt 0      (L(i) done, S(i-1) done)
//                      S(i) <- buf[i&1]       (overlaps with L(i+1))
// One ASYNCcnt drain per row instead of two; each row's HBM store overlaps
// the next row's L2 load. The 839 MB store stream is the roofline bound.
// ---------------------------------------------------------------------------
__global__ __launch_bounds__(512)
void gather_rows_kernel(const int* __restrict__ hours,
                        const float* __restrict__ P,
                        float* __restrict__ out,
                        int nrows) {
    __shared__ float buf[2][NUM_CATS];              // 2 x 8 KB staging buffers
    const int tid = threadIdx.x;                    // 0 .. 511
    const unsigned lds0 = lds_offset_of(&buf[0][tid * 4]);
    const unsigned lds1 = lds_offset_of(&buf[1][tid * 4]);

    int parity = 0;
    for (int row = blockIdx.x; row < nrows; row += gridDim.x) {
        const int hour = hours[row];                // uniform -> scalar load

        const char* gsrc = (const char*)(P + (size_t)hour * NUM_CATS) + tid * 16;
        char*       gdst = (char*)(out + (size_t)row * NUM_CATS) + tid * 16;
        const unsigned l = parity ? lds1 : lds0;

        async_load_b128_to_lds(l, gsrc);            // fills buf[parity]
        wait_asynccnt0();                           // load done; prev store done
        async_store_b128_from_lds(gdst, l);         // drains buf[parity] async

        parity ^= 1;                                // next row uses other buffer
    }
    // S_ENDPGM performs an implicit wait-idle; the final store completes
    // before the wave retires.
}

// ---------------------------------------------------------------------------
// Host launcher.
//   d_in[0]: inputs_hour        int32  [512*200]
//   d_in[1]: catid_time_matrix  f32    [168*2048]
//   d_out  : f32 [512*200*2048]
//   d_ws   : >= 168*2048*4 = 1.375 MB scratch for the softmaxed table
// ---------------------------------------------------------------------------
extern "C" void kernel_launch(void* const* d_in, const int* in_sizes, int n_in,
                              void* d_out, int out_size, void* d_ws, size_t ws_size,
                              hipStream_t stream) {
    const int*   hours = (const int*)d_in[0];
    const float* M     = (const float*)d_in[1];
    float*       out   = (float*)d_out;
    float*       probs = (float*)d_ws;

    const int nrows = in_sizes[0];                  // 512*200 = 102400

    softmax_rows_kernel<<<NUM_SLOTS, 256, 0, stream>>>(M, probs);

    int grid = 8192;
    if (grid > nrows) grid = nrows;
    gather_rows_kernel<<<grid, 512, 0, stream>>>(hours, probs, out, nrows);
}